// GCN_37941741093315
// MI455X (gfx1250) — compile-verified
//
#include <hip/hip_runtime.h>

typedef __attribute__((ext_vector_type(2))) float v2f;
typedef __attribute__((ext_vector_type(8))) float v8f;

#define WPB 8  // waves per block (256 threads, wave32)

// ---------------------------------------------------------------------------
// Dense GEMM Y[N, pad4(FOUT)] = X[N, pad4(FIN)] @ W[FIN, FOUT]
// via V_WMMA_F32_16X16X4_F32, one wave per 16x16 output tile.
//
// Compile-time FIN/FOUT  -> fully unrolled K loop, no runtime guards.
// X row stride = pad4(FIN): A loads are unconditional 8B global_load_b64.
// W staged zero-padded in LDS [KP][NP]: B loads are unconditional ds loads,
// and output columns >= FOUT compute to exact 0 (auto zero-padding of Y).
// N must be a multiple of 16 (100000 = 6250*16).
//
// Fragment layouts (wave32, f32):
//  A 16x4 : lane L -> m=L&15, kh=L>>4 ; a = {A[m][k0+2kh], A[m][k0+2kh+1]}
//  B 4x16 : lane L -> n=L&15, kh=L>>4 ; b = {B[k0+2kh][n], B[k0+2kh+1][n]}
//  C/D    : lane L -> n=L&15 ; VGPR i holds row (i + 8*kh)
// ---------------------------------------------------------------------------
template <int FIN, int FOUT>
__global__ void __launch_bounds__(256) gemm_wmma(
    const float* __restrict__ X, const float* __restrict__ W,
    float* __restrict__ Y, int N)
{
    constexpr int KP = (FIN  + 3) & ~3;     // padded K (X row stride)
    constexpr int TN = (FOUT + 15) / 16;    // 16-wide column tiles
    constexpr int NP = TN * 16;             // padded cols in LDS
    constexpr int YS = (FOUT + 3) & ~3;     // Y row stride

    __shared__ float Ws[KP * NP];

    // Cooperative zero-padded stage of W into LDS.
    for (int t = threadIdx.x; t < KP * NP; t += 256) {
        int k = t / NP;
        int c = t - k * NP;
        Ws[t] = (k < FIN && c < FOUT) ? W[k * FOUT + c] : 0.0f;
    }
    __syncthreads();

    int tilesM = N >> 4;
    int wave   = blockIdx.x * WPB + (threadIdx.x >> 5);
    if (wave >= tilesM * TN) return;        // wave-uniform exit (post-barrier)

    int lane  = threadIdx.x & 31;
    int tileM = wave / TN;
    int tileN = wave - tileM * TN;
    int m     = lane & 15;
    int kh    = lane >> 4;
    int row   = tileM * 16 + m;             // always < N
    int col   = tileN * 16 + m;

    const float* __restrict__ xrow = X + (size_t)row * KP;

    v8f acc = {};
    #pragma unroll
    for (int k0 = 0; k0 < KP; k0 += 4) {
        int ka = k0 + 2 * kh;               // even -> 8B-aligned
        v2f a  = *(const v2f*)(xrow + ka);
        v2f b;
        b.x = Ws[(ka    ) * NP + col];
        b.y = Ws[(ka + 1) * NP + col];
        // 8 args: (neg_a, A, neg_b, B, c_mod, C, reuse_a, reuse_b)
        acc = __builtin_amdgcn_wmma_f32_16x16x4_f32(
            false, a, false, b, (short)0, acc, false, false);
    }

    if (col < YS) {                         // pad cols < YS hold exact zeros
        #pragma unroll
        for (int i = 0; i < 8; ++i) {
            int r = tileM * 16 + i + 8 * kh;
            Y[(size_t)r * YS + col] = acc[i];
        }
    }
}

// ---------------------------------------------------------------------------
// Degree / normalization helpers
// ---------------------------------------------------------------------------
__global__ void init_one_k(float* __restrict__ p, int n) {
    int i = blockIdx.x * blockDim.x + threadIdx.x;
    if (i < n) p[i] = 1.0f;
}

__global__ void deg_add_k(const int* __restrict__ dst, float* __restrict__ deg, int e) {
    int i = blockIdx.x * blockDim.x + threadIdx.x;
    if (i < e) atomicAdd(&deg[dst[i]], 1.0f);
}

__global__ void rsqrt_k(float* __restrict__ p, int n) {
    int i = blockIdx.x * blockDim.x + threadIdx.x;
    if (i < n) p[i] = rsqrtf(p[i]);
}

__global__ void zero_k(float* __restrict__ p, long long n) {
    long long i = (long long)blockIdx.x * blockDim.x + threadIdx.x;
    if (i < n) p[i] = 0.0f;
}

// ---------------------------------------------------------------------------
// Edge scatter: agg[dst, f] += xw[src, f] * dinv[src] * dinv[dst]
// One thread per (edge, feature); S = padded row stride. Working set is
// L2-resident (<=95 MB total) so this runs at L2 atomic bandwidth.
// ---------------------------------------------------------------------------
__global__ void scatter_k(const float* __restrict__ xw,
                          const int* __restrict__ src, const int* __restrict__ dst,
                          const float* __restrict__ dinv, float* __restrict__ agg,
                          long long total, int Fout, int S)
{
    long long idx = (long long)blockIdx.x * blockDim.x + threadIdx.x;
    if (idx >= total) return;
    int e = (int)(idx / Fout);
    int f = (int)(idx - (long long)e * Fout);
    int s = src[e];
    int d = dst[e];
    float w = dinv[s] * dinv[d];
    atomicAdd(&agg[(size_t)d * S + f], xw[(size_t)s * S + f] * w);
}

// ---------------------------------------------------------------------------
// Combine: h = agg + xw*dinv^2 + b, optional ReLU, deterministic hash dropout.
// Operates over the padded [N, S] layout; pad columns are forced to zero so
// the buffer is a valid K-padded input for the next layer's GEMM.
// compact!=0 (last layer): write only f==0 densely into out[N].
// ---------------------------------------------------------------------------
__device__ __forceinline__ unsigned hash_u32(unsigned x) {
    x ^= x >> 16; x *= 0x7feb352du;
    x ^= x >> 15; x *= 0x846ca68bu;
    x ^= x >> 16; return x;
}

__global__ void combine_k(const float* __restrict__ xw, const float* __restrict__ agg,
                          const float* __restrict__ dinv, const float* __restrict__ bias,
                          float* __restrict__ out, long long total, int S, int Fout,
                          int doRelu, float dropP, unsigned seed, int compact)
{
    long long idx = (long long)blockIdx.x * blockDim.x + threadIdx.x;
    if (idx >= total) return;
    int i = (int)(idx / S);
    int f = (int)(idx - (long long)i * S);
    if (f < Fout) {
        float di = dinv[i];
        float v  = agg[idx] + xw[idx] * di * di + bias[f];
        if (doRelu) v = fmaxf(v, 0.0f);
        if (dropP > 0.0f) {
            unsigned h = hash_u32((unsigned)((long long)i * Fout + f) * 2654435761u + seed);
            float r = (float)(h >> 8) * (1.0f / 16777216.0f);
            v = (r >= dropP) ? v / (1.0f - dropP) : 0.0f;
        }
        if (compact) { if (f == 0) out[i] = v; }
        else         out[idx] = v;
    } else if (!compact) {
        out[idx] = 0.0f;                    // keep K-pad columns zero
    }
}

// ---------------------------------------------------------------------------
// Host orchestration. Input order: x, m, f, edge_index, W1,b1,W2,b2,W3,b3,
// W4,b4, WA,bA, WA1,bA1. Dead branch (WA/WA1) skipped — never reaches output.
// ---------------------------------------------------------------------------
static inline int pad4(int v) { return (v + 3) & ~3; }

extern "C" void kernel_launch(void* const* d_in, const int* in_sizes, int n_in,
                              void* d_out, int out_size, void* d_ws, size_t ws_size,
                              hipStream_t stream)
{
    const float* x  = (const float*)d_in[0];
    const int*   ei = (const int*)d_in[3];
    int N = in_sizes[0] / 32;
    int E = in_sizes[3] / 2;
    const int* src = ei;
    const int* dst = ei + E;

    const float* W[4]  = {(const float*)d_in[4], (const float*)d_in[6],
                          (const float*)d_in[8], (const float*)d_in[10]};
    const float* Bv[4] = {(const float*)d_in[5], (const float*)d_in[7],
                          (const float*)d_in[9], (const float*)d_in[11]};
    const int   fout[4]  = {100, 50, 20, 1};
    const float dropP[4] = {0.2f, 0.1f, 0.1f, 0.0f};

    // Workspace: dinv[N] | bufA[N*112] | bufB[N*112]  (~90 MB)
    float* dinv = (float*)d_ws;
    float* bufA = dinv + N;
    float* bufB = bufA + (size_t)N * 112;

    const int T = 256;

    // deg = 1 + histogram(dst); dinv = rsqrt(deg) in place
    init_one_k<<<(N + T - 1) / T, T, 0, stream>>>(dinv, N);
    deg_add_k<<<(E + T - 1) / T, T, 0, stream>>>(dst, dinv, E);
    rsqrt_k<<<(N + T - 1) / T, T, 0, stream>>>(dinv, N);

    const float* in = x;
    for (int l = 0; l < 4; ++l) {
        int Fout = fout[l];
        int S    = pad4(Fout);
        int TNl  = (Fout + 15) / 16;
        int waves   = (N / 16) * TNl;
        int gBlocks = (waves + WPB - 1) / WPB;

        // 1) xw = in @ W -> bufA (WMMA f32, compile-time dims)
        switch (l) {
            case 0: gemm_wmma< 32, 100><<<gBlocks, T, 0, stream>>>(in, W[l], bufA, N); break;
            case 1: gemm_wmma<100,  50><<<gBlocks, T, 0, stream>>>(in, W[l], bufA, N); break;
            case 2: gemm_wmma< 50,  20><<<gBlocks, T, 0, stream>>>(in, W[l], bufA, N); break;
            case 3: gemm_wmma< 20,   1><<<gBlocks, T, 0, stream>>>(in, W[l], bufA, N); break;
        }

        // 2) zero agg buffer (input buffer is dead after the GEMM)
        long long ns = (long long)N * S;
        zero_k<<<(int)((ns + T - 1) / T), T, 0, stream>>>(bufB, ns);

        // 3) edge scatter with symmetric normalization (atomic, L2-resident)
        long long ef = (long long)E * Fout;
        scatter_k<<<(int)((ef + T - 1) / T), T, 0, stream>>>(bufA, src, dst, dinv,
                                                             bufB, ef, Fout, S);

        // 4) self-loop + bias + relu + dropout (in place; last layer -> d_out)
        float* outp = (l == 3) ? (float*)d_out : bufB;
        combine_k<<<(int)((ns + T - 1) / T), T, 0, stream>>>(
            bufA, bufB, dinv, Bv[l], outp, ns, S, Fout,
            (l < 3) ? 1 : 0, dropP[l], 0x9E3779B9u + (unsigned)l, (l == 3) ? 1 : 0);

        in = bufB;
    }
}